// HypeEntropyModelSoS_45157286150659
// MI455X (gfx1250) — compile-verified
//
#include <hip/hip_runtime.h>

// ---------------------------------------------------------------------------
// HypeEntropyModelSoS quantize+dequantize for MI455X (gfx1250, wave32).
//
// Memory-bound elementwise kernel: 16 B of HBM traffic per element
// (read x, means; write symbol, dequant) -> ~75.5 MB @ 23.3 TB/s ~ 3.2 us.
// Compute is a 6-step branchless binary search over a 64-entry sorted
// codebook held in LDS. WMMA is deliberately NOT used: there is no
// contraction; a matmul reformulation (-2zc + c^2) saves no bytes and adds
// 64x arithmetic. CDNA5-specific paths used instead:
//   * global_load_async_to_lds_b128 + s_wait_asynccnt (ASYNCcnt DMA) to
//     broadcast the 256-byte codebook into LDS once per workgroup
//   * non-temporal 128-bit global loads/stores (streams touched once)
//   * ds_load-based branchless search (wave32, zero divergence)
// ---------------------------------------------------------------------------

typedef float v4f __attribute__((ext_vector_type(4)));
typedef int   v4i __attribute__((ext_vector_type(4)));

#define NLEV 64

__global__ __launch_bounds__(256) void sos_quant_kernel(
    const float* __restrict__ x,
    const float* __restrict__ means,
    const float* __restrict__ codebook,
    int*   __restrict__ sym,
    float* __restrict__ deq,
    int n4)   // number of float4 groups
{
    __shared__ float cb[NLEV];

    const int tid = threadIdx.x;

    // --- Async DMA: 16 lanes x 16 B = 256 B codebook -> LDS ---------------
    if (tid < NLEV / 4) {
        // Low 32 bits of the generic pointer to a __shared__ object are the
        // LDS byte offset (shared-aperture flat addressing, ISA 10.2).
        unsigned lds_off = (unsigned)(size_t)(&cb[tid * 4]);
        unsigned long long gaddr =
            (unsigned long long)(const void*)(codebook + tid * 4);
        asm volatile("global_load_async_to_lds_b128 %0, %1, off"
                     :: "v"(lds_off), "v"(gaddr)
                     : "memory");
    }
    // Issuing wave drains its ASYNCcnt; barrier publishes LDS to all waves.
    asm volatile("s_wait_asynccnt 0" ::: "memory");
    __syncthreads();

    long long gid = (long long)blockIdx.x * blockDim.x + tid;
    if (gid >= n4) return;

    // Streaming 128-bit non-temporal loads (wave32: 512 B contiguous/wave).
    v4f xv = __builtin_nontemporal_load((const v4f*)x + gid);
    v4f mv = __builtin_nontemporal_load((const v4f*)means + gid);

    v4i s;
    v4f d;
#pragma unroll
    for (int j = 0; j < 4; ++j) {
        float z = xv[j] - mv[j];

        // Branchless lower-bound over sorted cb: idx = max{i : cb[i] <= z},
        // or 0 if z < cb[0]. Six v_cndmask-selected ds_load_b32 steps.
        int idx = 0;
        idx += (cb[idx + 32] <= z) ? 32 : 0;
        idx += (cb[idx + 16] <= z) ? 16 : 0;
        idx += (cb[idx +  8] <= z) ?  8 : 0;
        idx += (cb[idx +  4] <= z) ?  4 : 0;
        idx += (cb[idx +  2] <= z) ?  2 : 0;
        idx += (cb[idx +  1] <= z) ?  1 : 0;

        int hi = (idx < NLEV - 1) ? idx + 1 : NLEV - 1;
        float clo = cb[idx];
        float chi = cb[hi];
        // Strict '<' => ties pick the lower index (jnp.argmin first-min).
        bool up = fabsf(z - chi) < fabsf(z - clo);
        s[j] = up ? hi : idx;
        d[j] = (up ? chi : clo) + mv[j];
    }

    __builtin_nontemporal_store(s, (v4i*)sym + gid);
    __builtin_nontemporal_store(d, (v4f*)deq + gid);
}

extern "C" void kernel_launch(void* const* d_in, const int* in_sizes, int n_in,
                              void* d_out, int out_size, void* d_ws, size_t ws_size,
                              hipStream_t stream) {
    const float* x     = (const float*)d_in[0];
    const float* means = (const float*)d_in[1];
    const float* cbook = (const float*)d_in[2];

    const int n  = in_sizes[0];          // 4*192*64*96 = 4,718,592 (divisible by 4)
    const int n4 = n / 4;

    int*   sym = (int*)d_out;            // symbols: int32 bit patterns, first n slots
    float* deq = (float*)d_out + n;      // dequant: next n floats

    const int threads = 256;             // 8 wave32 waves per block
    const int blocks  = (n4 + threads - 1) / threads;   // 4608 blocks

    sos_quant_kernel<<<blocks, threads, 0, stream>>>(x, means, cbook, sym, deq, n4);
}